// ModuleFormerAttention_19215683682718
// MI455X (gfx1250) — compile-verified
//
#include <hip/hip_runtime.h>

// CDNA5 / gfx1250 stick-breaking (ModuleFormer) attention, fused flash-style.
// fp32 WMMA 16x16x4 for both QK^T and P*V; reverse-j streaming for the
// exclusive reverse cumsum of log_beta (log2 domain, DPP suffix scan);
// async-to-LDS K (b128 row-major) and V (b32 transpose scatter), ping-pong
// buffers resolved at compile time via 2x loop unroll.

typedef __attribute__((ext_vector_type(2))) float v2f;
typedef __attribute__((ext_vector_type(8))) float v8f;

#define T_SEQ 2048
#define KTOP  2
#define NH    8
#define DH    64
#define ROWS_PER_WG 128        // 8 waves x 16 rows
#define KPAD  68               // LDS row stride for K tile (conflict-free, 16B aligned)
#define VTP   18               // LDS row stride for transposed V tile Vt[d][j]
#define PPAD  18               // LDS row stride for per-wave P tile (8B aligned pairs)
#define SCALE_LOG2E 0.180336880f   // (1/sqrt(64)) * log2(e)
#define JBLK_BYTES (16ull * NH * DH * 4ull)   // global bytes between j-blocks (32 KB)

// suffix-shift within each 16-lane DPP row, zero fill (old = 0)
#define DPP_SHL_Z(S, CTRL) \
    __int_as_float(__builtin_amdgcn_update_dpp(0, __float_as_int(S), (CTRL), 0xf, 0xf, false))

#define ASYNC_WAIT_SYNC() do {                                   \
    asm volatile("s_wait_asynccnt 0x0" ::: "memory");            \
    __syncthreads();                                             \
} while (0)

__device__ __forceinline__ float row_bcast0(float s) {
#if __has_builtin(__builtin_amdgcn_permlane16)
    return __int_as_float(__builtin_amdgcn_permlane16(__float_as_int(s), __float_as_int(s),
                                                      0u, 0u, false, false));
#else
    return __shfl(s, 0, 16);
#endif
}

// Pointwise stick-breaking weights in log2 domain.
//   t   = logits * log2(e)
//   lb2 = log2(sigmoid(-x)) = -(max(t,0) + log2(1 + 2^-|t|))
//   s2  = inclusive suffix scan of lb2 over the 16 lanes holding the row
//   p   = sigmoid(x)*exp(excl_suffix + carry) = 2^(t + s2 + carry2)
template<bool MASKED>
__device__ __forceinline__ void sb_pointwise(const v8f S, int jb, int i0u,
                                             int l16, int half,
                                             float* accrow, float* pw)
{
#pragma unroll
    for (int g = 0; g < 8; ++g) {
        float t = S[g] * SCALE_LOG2E;
        bool keep = true;
        if (MASKED) keep = (jb * 16 + l16) <= (i0u + g + 8 * half);
        float e   = __builtin_amdgcn_exp2f(-fabsf(t));
        float lb2 = -fmaxf(t, 0.0f) - __builtin_amdgcn_logf(1.0f + e);
        if (MASKED && !keep) lb2 = 0.0f;
        float s = lb2;                       // inclusive suffix sum (row_shl, zero fill)
        s += DPP_SHL_Z(s, 0x101);            // += lane+1
        s += DPP_SHL_Z(s, 0x102);            // += lane+2
        s += DPP_SHL_Z(s, 0x104);            // += lane+4
        s += DPP_SHL_Z(s, 0x108);            // += lane+8
        float p = __builtin_amdgcn_exp2f(t + (s + accrow[g]));
        if (MASKED && !keep) p = 0.0f;
        accrow[g] += row_bcast0(s);          // row total = suffix at col 0
        pw[(g + 8 * half) * PPAD + l16] = p;
    }
}

__global__ __launch_bounds__(256)
void sb_attn_f32wmma_kernel(const float* __restrict__ q,
                            const float* __restrict__ kmat,
                            const float* __restrict__ vmat,
                            float* __restrict__ y)
{
    __shared__ float Ksh[2][16 * KPAD];        // ping-pong K block: 16(j) x 64(d)
    __shared__ float Vts[2][DH * VTP];         // ping-pong transposed V: 64(d) x 16(j)
    __shared__ float Psh[8 * 16 * PPAD];       // per-wave P tile (C-layout -> A-layout bounce)

    const int tid  = threadIdx.x;
    const int wave = tid >> 5;
    const int lane = tid & 31;
    const int l16  = lane & 15;
    const int half = lane >> 4;

    const int tile = blockIdx.x;               // which 128-row tile
    const int kh   = blockIdx.y;               // kk*NH + h
    const int kk   = kh >> 3;
    const int h    = kh & 7;

    const int i0w = tile * ROWS_PER_WG + wave * 16;      // first row this wave owns
    const int i0u = __builtin_amdgcn_readfirstlane(i0w); // provably wave-uniform copy

    const int jb_max = tile * 8 + 7;           // diagonal block (odd; trip count even)

    // ---- async staging state ---------------------------------------------
    // K: one b128 per lane (row srow, d-quad sc4). V: 4 b32 per lane, scattered
    // into the transpose Vt[d][j]. Global addresses are running pointers,
    // stepped down by one j-block (32 KB) per issue.
    const int srow = tid >> 4;                 // 0..15  (j within block)
    const int sc4  = (tid & 15) << 2;          // 0,4,...,60 (d quad)  [K path]
    const int dv   = tid & 15;                 // base d               [V path]

    const size_t grow0 = (size_t)(jb_max * 16 + srow) * (NH * DH) + (size_t)h * DH;
    unsigned long long kga = (unsigned long long)(size_t)(kmat + grow0 + sc4);
    unsigned long long vga = (unsigned long long)(size_t)(vmat + grow0 + dv);

    unsigned kdst[2], vdst[2][4];
#pragma unroll
    for (int b = 0; b < 2; ++b) {
        kdst[b] = (unsigned)(size_t)&Ksh[b][srow * KPAD + sc4];
#pragma unroll
        for (int r = 0; r < 4; ++r)
            vdst[b][r] = (unsigned)(size_t)&Vts[b][(dv + 16 * r) * VTP + srow];
    }

    auto issue_stage = [&](int b) {
        asm volatile("global_load_async_to_lds_b128 %0, %1, off"
                     :: "v"(kdst[b]), "v"(kga) : "memory");
#pragma unroll
        for (int r = 0; r < 4; ++r)
            asm volatile("global_load_async_to_lds_b32 %0, %1, off"
                         :: "v"(vdst[b][r]), "v"(vga + 64ull * r) : "memory");
        kga -= JBLK_BYTES;
        vga -= JBLK_BYTES;
    };

    // ---- Q tile (16x64) into A-operand registers -------------------------
    // A chunk c covers d = 4c..4c+3 ; element(vgpr r, lane L) = Q[L&15][4c + r + 2*(L>>4)]
    v2f qa[16];
    {
        const size_t qbase = (size_t)(i0w + l16) * (KTOP * NH * DH)
                           + (size_t)kk * (NH * DH) + (size_t)h * DH;
#pragma unroll
        for (int c = 0; c < 16; ++c)
            qa[c] = *(const v2f*)(q + qbase + 4 * c + 2 * half);
    }

    // output accumulators: four 16(i)x16(d) f32 C tiles
    v8f acc[4];
#pragma unroll
    for (int dt = 0; dt < 4; ++dt) acc[dt] = (v8f)0.0f;

    // running log2-domain suffix carry per owned row (C-layout: row = g + 8*half)
    float accrow[8];
#pragma unroll
    for (int g = 0; g < 8; ++g) accrow[g] = 0.0f;

    float* pw = &Psh[wave * 16 * PPAD];

    // one j-block worth of work against compile-time buffer b
    auto compute = [&](int jb, const float* Kc, const float* Vt) {
        // blocks fully above this wave's rows contribute exactly zero: skip.
        if (jb * 16 > i0u + 15) return;

        // ---- GEMM1: S(16x16) = Q_tile * K_block^T ----
        v8f S = (v8f)0.0f;
#pragma unroll
        for (int c = 0; c < 16; ++c) {
            v2f kb = *(const v2f*)&Kc[l16 * KPAD + 4 * c + 2 * half];
            S = __builtin_amdgcn_wmma_f32_16x16x4_f32(false, qa[c], false, kb,
                                                      (short)0, S, false, false);
        }

        // ---- pointwise: masked only for the single diagonal-band block ----
        if (jb * 16 + 15 > i0u)
            sb_pointwise<true>(S, jb, i0u, l16, half, accrow, pw);
        else
            sb_pointwise<false>(S, jb, i0u, l16, half, accrow, pw);

        // ---- GEMM2: Y(16x64) += P(16x16) * V_block(16x64) ----
#pragma unroll
        for (int j0 = 0; j0 < 16; j0 += 4) {
            // A chunk: element(r, L) = P[L&15][j0 + r + 2*(L>>4)]
            v2f pa = *(const v2f*)&pw[l16 * PPAD + j0 + 2 * half];
            const int jr = j0 + 2 * half;
#pragma unroll
            for (int dt = 0; dt < 4; ++dt) {
                // B chunk: element(r, L) = V[jr + r][dt*16 + (L&15)]
                //        = Vt[dt*16 + l16][jr + r]  -> contiguous b64
                v2f vb = *(const v2f*)&Vt[(dt * 16 + l16) * VTP + jr];
                acc[dt] = __builtin_amdgcn_wmma_f32_16x16x4_f32(false, pa, false, vb,
                                                                (short)0, acc[dt],
                                                                false, false);
            }
        }
    };

    // ---- main loop: 2x unrolled, compile-time ping-pong ------------------
    issue_stage(0);                            // stage jb_max into buffer 0
    ASYNC_WAIT_SYNC();

    for (int jb = jb_max; jb >= 0; jb -= 2) {
        if (jb > 0) issue_stage(1);            // prefetch jb-1 -> buf 1
        compute(jb, Ksh[0], Vts[0]);
        ASYNC_WAIT_SYNC();

        if (jb > 1) issue_stage(0);            // prefetch jb-2 -> buf 0
        compute(jb - 1, Ksh[1], Vts[1]);
        ASYNC_WAIT_SYNC();
    }

    // ---- store Y tile: C-layout -> y[i, kk, h, d] -------------------------
#pragma unroll
    for (int g = 0; g < 8; ++g) {
        const size_t yb = (size_t)(i0w + g + 8 * half) * (KTOP * NH * DH)
                        + (size_t)kk * (NH * DH) + (size_t)h * DH + l16;
#pragma unroll
        for (int dt = 0; dt < 4; ++dt)
            y[yb + dt * 16] = acc[dt][g];
    }
}

extern "C" void kernel_launch(void* const* d_in, const int* in_sizes, int n_in,
                              void* d_out, int out_size, void* d_ws, size_t ws_size,
                              hipStream_t stream) {
    (void)in_sizes; (void)n_in; (void)d_ws; (void)ws_size; (void)out_size;
    const float* q = (const float*)d_in[0];
    const float* k = (const float*)d_in[1];
    const float* v = (const float*)d_in[2];
    float* y = (float*)d_out;

    dim3 grid(T_SEQ / ROWS_PER_WG, KTOP * NH);   // (16 row tiles, 16 (k,h) pairs)
    dim3 block(256);                              // 8 waves (wave32)
    sb_attn_f32wmma_kernel<<<grid, block, 0, stream>>>(q, k, v, y);
}